// CapsuleLayer_39505109189132
// MI455X (gfx1250) — compile-verified
//
#include <hip/hip_runtime.h>

typedef __attribute__((ext_vector_type(2))) float v2f;
typedef __attribute__((ext_vector_type(8))) float v8f;

#define L_SEQ   200
#define D_DIM   64
#define O_DIM   64
#define K_CAPS  3
#define L_PAD   208      // 13 strips * 16
#define M_STRIDE 66      // padded LDS row stride for mappings (bank-conflict free)
#define NEG_BIG (-1e9f)

__launch_bounds__(256)
__global__ void capsule_fused_kernel(const float* __restrict__ E,
                                     const float* __restrict__ Wg,
                                     const float* __restrict__ logits0,
                                     const int*   __restrict__ seqlen,
                                     float*       __restrict__ out) {
    __shared__ float MLDS[L_PAD * M_STRIDE];   // mappings [208 x 64], stride 66
    __shared__ float WL[D_DIM * O_DIM];        // staged W [64 x 64]
    __shared__ float wLDS[K_CAPS * L_PAD];     // softmax weights
    __shared__ float candLDS[K_CAPS * O_DIM];  // candidates [3 x 64]
    __shared__ float factLDS[K_CAPS];          // squash factors

    const int tid  = threadIdx.x;
    const int n    = blockIdx.x;
    const int lane = tid & 31;
    const int wave = tid >> 5;
    const int m15  = lane & 15;   // M index within tile (all lanes)
    const int hi   = lane >> 4;   // 0: K pair {0,1}, 1: K pair {2,3}

    // ---- stage W into LDS (flat, coalesced) ----
    #pragma unroll
    for (int i = 0; i < 16; ++i)
        WL[tid + 256 * i] = Wg[tid + 256 * i];
    __syncthreads();

    // ---- preload B fragments (strip-invariant) into registers ----
    // B 4x16 f32 layout: VGPR0 = rows {kb, kb+2} (lane halves), VGPR1 = rows {kb+1, kb+3}
    v2f bfr[16][4];
    #pragma unroll
    for (int kk = 0; kk < 16; ++kk) {
        const int kb = kk * 4 + 2 * hi;
        #pragma unroll
        for (int nt = 0; nt < 4; ++nt) {
            v2f b;
            b.x = WL[(kb    ) * O_DIM + nt * 16 + m15];
            b.y = WL[(kb + 1) * O_DIM + nt * 16 + m15];
            bfr[kk][nt] = b;
        }
    }

    const float* En = E + (size_t)n * (L_SEQ * D_DIM);

    // ---- mappings = E[n] (200x64) @ W (64x64) via V_WMMA_F32_16X16X4_F32 ----
    // 13 M-strips over 8 waves; each wave holds 4 N-tile accumulators.
    for (int strip = wave; strip < 13; strip += 8) {
        const int  row   = strip * 16 + m15;
        const bool rowok = (row < L_SEQ);
        // clamp pad rows (200..207) to row 0 so loads are unconditional,
        // then zero those lanes' fragments below.
        const float* Erow = En + (size_t)(rowok ? row : 0) * D_DIM;

        // batch-issue all 16 A-fragment loads (non-temporal: E has no reuse)
        v2f afr[16];
        #pragma unroll
        for (int kk = 0; kk < 16; ++kk)
            afr[kk] = __builtin_nontemporal_load(
                (const v2f*)(Erow + kk * 4 + 2 * hi));
        if (!rowok) {
            #pragma unroll
            for (int kk = 0; kk < 16; ++kk) { afr[kk].x = 0.0f; afr[kk].y = 0.0f; }
        }

        v8f zero = {};
        v8f acc[4] = {zero, zero, zero, zero};
        #pragma unroll
        for (int kk = 0; kk < 16; ++kk) {
            #pragma unroll
            for (int nt = 0; nt < 4; ++nt) {
                acc[nt] = __builtin_amdgcn_wmma_f32_16x16x4_f32(
                    false, afr[kk], false, bfr[kk][nt], (short)0, acc[nt],
                    false, false);
            }
        }

        // C/D layout: VGPR v -> rows v (lanes 0-15) and v+8 (lanes 16-31)
        #pragma unroll
        for (int nt = 0; nt < 4; ++nt)
            #pragma unroll
            for (int v = 0; v < 8; ++v)
                MLDS[(strip * 16 + v + 8 * hi) * M_STRIDE + nt * 16 + m15] =
                    acc[nt][v];
    }

    // ---- routing (all on-chip) ----
    const int  slen   = seqlen[n];
    const int  l      = tid;
    const bool lvalid = (l < L_SEQ);
    const bool mvalid = lvalid && (l < slen);

    float lg0 = 0.f, lg1 = 0.f, lg2 = 0.f;
    if (lvalid) {
        const float* lp = logits0 + (size_t)n * (K_CAPS * L_SEQ);
        lg0 = lp[0 * L_SEQ + l];
        lg1 = lp[1 * L_SEQ + l];
        lg2 = lp[2 * L_SEQ + l];
    }

    for (int it = 0; it < 3; ++it) {
        // softmax over capsule dim (masked positions -> uniform 1/3)
        if (lvalid) {
            float p0 = mvalid ? lg0 : NEG_BIG;
            float p1 = mvalid ? lg1 : NEG_BIG;
            float p2 = mvalid ? lg2 : NEG_BIG;
            float m  = fmaxf(p0, fmaxf(p1, p2));
            float e0 = __expf(p0 - m);
            float e1 = __expf(p1 - m);
            float e2 = __expf(p2 - m);
            float inv = 1.0f / (e0 + e1 + e2);
            wLDS[0 * L_PAD + l] = e0 * inv;
            wLDS[1 * L_PAD + l] = e1 * inv;
            wLDS[2 * L_PAD + l] = e2 * inv;
        }
        __syncthreads();  // also covers GEMM MLDS stores on it==0

        // candidates[k][o] = sum_l w[k][l] * M[l][o]  (all l, incl. masked)
        if (tid < K_CAPS * O_DIM) {
            const int k = tid >> 6, o = tid & 63;
            float acc = 0.0f;
            #pragma unroll 8
            for (int ll = 0; ll < L_SEQ; ++ll)
                acc += wLDS[k * L_PAD + ll] * MLDS[ll * M_STRIDE + o];
            candLDS[k * O_DIM + o] = acc;
        }
        __syncthreads();

        // squash factor per capsule
        if (tid < K_CAPS) {
            float sn = 0.0f;
            #pragma unroll 8
            for (int o = 0; o < O_DIM; ++o) {
                float c = candLDS[tid * O_DIM + o];
                sn += c * c;
            }
            factLDS[tid] = sn / (1.0f + sn) / sqrtf(sn + 1e-8f);
        }
        __syncthreads();

        if (it < 2) {
            // logits[k][l] += dot(interests[k], M[l])
            if (lvalid) {
                float d0 = 0.f, d1 = 0.f, d2 = 0.f;
                const float* mrow = &MLDS[l * M_STRIDE];
                #pragma unroll 8
                for (int o = 0; o < O_DIM; ++o) {
                    float mv = mrow[o];
                    d0 += candLDS[0 * O_DIM + o] * mv;
                    d1 += candLDS[1 * O_DIM + o] * mv;
                    d2 += candLDS[2 * O_DIM + o] * mv;
                }
                lg0 += factLDS[0] * d0;
                lg1 += factLDS[1] * d1;
                lg2 += factLDS[2] * d2;
            }
        } else {
            // final interests = factor * candidates -> out[n][k][o]
            if (tid < K_CAPS * O_DIM) {
                const int k = tid >> 6;
                out[(size_t)n * (K_CAPS * O_DIM) + tid] =
                    factLDS[k] * candLDS[tid];
            }
        }
    }
}

extern "C" void kernel_launch(void* const* d_in, const int* in_sizes, int n_in,
                              void* d_out, int out_size, void* d_ws, size_t ws_size,
                              hipStream_t stream) {
    const float* E       = (const float*)d_in[0];  // [N, 200, 64]
    const float* W       = (const float*)d_in[1];  // [64, 64]
    const float* logits0 = (const float*)d_in[2];  // [N, 3, 200]
    const int*   seqlen  = (const int*)d_in[3];    // [N, 1]
    float*       out     = (float*)d_out;          // [N, 3, 64]

    const int N = in_sizes[0] / (L_SEQ * D_DIM);   // 4096
    capsule_fused_kernel<<<N, 256, 0, stream>>>(E, W, logits0, seqlen, out);
}